// ChunkBySlices_1700807049706
// MI455X (gfx1250) — compile-verified
//
#include <hip/hip_runtime.h>
#include <cstdint>
#include <cstddef>

// Problem constants (from the reference): x is (N, T, F) fp32.
constexpr int N_DIM = 64;
constexpr int T_DIM = 2048;
constexpr int F_DIM = 256;                  // 1024 bytes per row
constexpr int ROW_BYTES = F_DIM * 4;

constexpr int ROWS_PER_WAVE  = 8;           // 8 KB of async DMA in flight per wave per round-trip
constexpr int WAVES_PER_BLOCK = 8;          // 256 threads
constexpr int ROWS_PER_BLOCK = ROWS_PER_WAVE * WAVES_PER_BLOCK;  // 64 rows -> 64 KB LDS/block

// ---- CDNA5 async global<->LDS DMA (ASYNCcnt) helpers -----------------------
// ISA ch.10 / CDNA5 ISA: GV mode, VDST/VSRC = LDS byte address VGPR,
// VADDR = 64-bit global address VGPR pair. Tracked with ASYNCcnt.

__device__ __forceinline__ unsigned lds_offset(const void* p) {
  // Generic LDS pointer: addr[31:0] is the LDS byte offset (aperture layout).
  return (unsigned)(unsigned long long)(uintptr_t)p;
}

__device__ __forceinline__ void async_load_b128(unsigned lds, const float* g) {
  asm volatile("global_load_async_to_lds_b128 %0, %1, off"
               :: "v"(lds), "v"((unsigned long long)(uintptr_t)g)
               : "memory");
}

__device__ __forceinline__ void async_store_b128(float* g, unsigned lds) {
  asm volatile("global_store_async_from_lds_b128 %0, %1, off"
               :: "v"((unsigned long long)(uintptr_t)g), "v"(lds)
               : "memory");
}

__device__ __forceinline__ void wait_async0() {
  asm volatile("s_wait_asynccnt 0x0" ::: "memory");
}

// ---- main kernel: one wave copies 8 consecutive (n,t) rows -----------------
// Block 0 additionally writes the chunk_lens tail (64 floats) so only one
// kernel launch is needed.
__global__ __launch_bounds__(WAVES_PER_BLOCK * 32)
void chunk_rows_kernel(const float* __restrict__ x,
                       const int*   __restrict__ slices,
                       const int*   __restrict__ lens,
                       float*       __restrict__ out) {
  __shared__ __align__(16) unsigned char smem[ROWS_PER_BLOCK * ROW_BYTES];

  const int wid  = (int)(threadIdx.x >> 5);
  const int lane = (int)(threadIdx.x & 31);

  // Fold the tiny chunk_lens output into block 0 (overlapped with the copy).
  if (blockIdx.x == 0 && threadIdx.x < N_DIM) {
    const int m  = (int)threadIdx.x;
    const int cl = max(slices[2 * m + 1] - slices[2 * m], 0);
    out[(size_t)N_DIM * T_DIM * F_DIM + m] = (float)cl;
  }

  // 64 rows per block, 8 per wave; T divisible by 8 -> a wave's batch never
  // crosses an n boundary, so the per-n scalars are batch-uniform.
  const int rowBase = (int)blockIdx.x * ROWS_PER_BLOCK + wid * ROWS_PER_WAVE;
  const int n  = rowBase / T_DIM;
  const int t0 = rowBase % T_DIM;

  const int start = slices[2 * n];
  const int end   = slices[2 * n + 1];
  const int chunk_len = max(end - start, 0);
  const int left_pad  = (chunk_len == 0) ? 0 : max(-start, 0);
  const int start_c   = max(start, 0);
  const int end_c     = min(end, lens[n]);
  const int slice_len = max(end_c - start_c, 0);
  const int vlo = left_pad;                 // valid window: [vlo, vhi)
  const int vhi = left_pad + slice_len;

  const unsigned ldsWave =
      lds_offset((const void*)smem) + (unsigned)(wid * (ROWS_PER_WAVE * ROW_BYTES));

  // Phase 1: fire async DMA loads (global -> LDS) for all valid rows.
#pragma unroll
  for (int j = 0; j < ROWS_PER_WAVE; ++j) {
    const int t = t0 + j;
    if (t >= vlo && t < vhi) {              // wave-uniform predicate
      const int src = start_c + t - left_pad;   // guaranteed in [0, T)
      const float* g = x + ((size_t)n * T_DIM + src) * F_DIM + lane * 4;
      const unsigned l = ldsWave + (unsigned)(j * ROW_BYTES + lane * 16);
      async_load_b128(l,        g);         // bytes [0,512) of the row
      async_load_b128(l + 512u, g + 128);   // bytes [512,1024)
    }
  }
  wait_async0();                            // all LDS tiles resident

  // Phase 2: async DMA stores (LDS -> global) for valid rows, zeros otherwise.
#pragma unroll
  for (int j = 0; j < ROWS_PER_WAVE; ++j) {
    const int t = t0 + j;
    float* o = out + ((size_t)n * T_DIM + t) * F_DIM + lane * 4;
    if (t >= vlo && t < vhi) {
      const unsigned l = ldsWave + (unsigned)(j * ROW_BYTES + lane * 16);
      async_store_b128(o,       l);
      async_store_b128(o + 128, l + 512u);
    } else {
      const float4 z = make_float4(0.f, 0.f, 0.f, 0.f);
      *(float4*)(o)       = z;              // global_store_b128
      *(float4*)(o + 128) = z;
    }
  }
  wait_async0();                            // drain before LDS/wave retire
}

extern "C" void kernel_launch(void* const* d_in, const int* in_sizes, int n_in,
                              void* d_out, int out_size, void* d_ws, size_t ws_size,
                              hipStream_t stream) {
  (void)in_sizes; (void)n_in; (void)out_size; (void)d_ws; (void)ws_size;

  const float* x      = (const float*)d_in[0];   // (N, T, F) fp32
  const int*   slices = (const int*)d_in[1];     // (N, 2) int
  const int*   lens   = (const int*)d_in[2];     // (N,) int
  float*       out    = (float*)d_out;           // chunks (N*T*F) ++ chunk_lens (N)

  const int nRowBlocks = (N_DIM * T_DIM) / ROWS_PER_BLOCK;   // 2048
  chunk_rows_kernel<<<nRowBlocks, WAVES_PER_BLOCK * 32, 0, stream>>>(x, slices, lens, out);
}